// CodeCloud_46969762349677
// MI455X (gfx1250) — compile-verified
//
#include <hip/hip_runtime.h>

// Problem constants (from the reference): B=1, P=65536, N=8192, D=128, K=32
#define P_TOTAL 65536
#define N_CODES 8192
#define DIMS    128
#define KNN     32
#define EPSF    1e-16f

typedef float v2f __attribute__((ext_vector_type(2)));
typedef float v8f __attribute__((ext_vector_type(8)));

// Block = 128 threads = 4 wave32s. Each wave owns 32 queries (two 16-query
// WMMA sub-tiles) and is fully independent: all LDS state is wave-private,
// so no workgroup barriers are needed (same-wave LDS ops are in-order on
// CDNA5; __builtin_amdgcn_wave_barrier() only pins compiler ordering).
__global__ __launch_bounds__(128) void CodeCloud_knn_wmma_kernel(
    const float* __restrict__ qpos,   // P x 3
    const float* __restrict__ cpos,   // N x 3
    const float* __restrict__ codes,  // N x D
    float* __restrict__ out)          // P x D
{
    // Per-wave d^2 tile: [wave][query 0..31][code 0..31 (+1 pad: 33 % 64 banks
    // coprime -> conflict-free per-lane row scans)]
    __shared__ float tileD[4][32][33];            // ~16.9 KB
    // Per-lane top-K lists (same padding)
    __shared__ float listD[4][32][33];            // ~16.9 KB
    __shared__ int   listI[4][32][33];            // ~16.9 KB

    const int wave  = threadIdx.x >> 5;
    const int lane  = threadIdx.x & 31;
    const int half  = lane >> 4;                  // 0: lanes 0-15, 1: lanes 16-31
    const int l16   = lane & 15;
    const int qbase = blockIdx.x * 128 + wave * 32;

    // ---- init per-lane top-K list ----
    #pragma unroll
    for (int k = 0; k < KNN; ++k) {
        listD[wave][lane][k] = __builtin_inff();
        listI[wave][lane][k] = 0;
    }
    float curMax = __builtin_inff();
    int   curMaxPos = 0;

    // ---- build the two A fragments (16x4 f32 A-matrix layout) ----
    // lanes 0-15 hold (K=0,K=1) = (qx,qy); lanes 16-31 hold (K=2,K=3) = (qz,|q|^2)
    v2f a0, a1;
    {
        int q0 = qbase + l16;
        float x = qpos[3*q0+0], y = qpos[3*q0+1], z = qpos[3*q0+2];
        float n2 = x*x + y*y + z*z;
        a0.x = half ? z  : x;
        a0.y = half ? n2 : y;
        int q1 = qbase + 16 + l16;
        x = qpos[3*q1+0]; y = qpos[3*q1+1]; z = qpos[3*q1+2];
        n2 = x*x + y*y + z*z;
        a1.x = half ? z  : x;
        a1.y = half ? n2 : y;
    }

    // ---- software-pipelined preload of the first 32-code tile ----
    float cAx = cpos[3*l16+0],      cAy = cpos[3*l16+1],      cAz = cpos[3*l16+2];
    float cBx = cpos[3*(16+l16)+0], cBy = cpos[3*(16+l16)+1], cBz = cpos[3*(16+l16)+2];

    // ---- main loop: 32 codes / iteration -> 4 WMMAs (one 32x32 d^2 tile) ----
    for (int nbase = 0; nbase < N_CODES; nbase += 32) {
        // Issue next iteration's cpos loads now (wrapped index: always valid,
        // no divergent guard); the WMMA + selection scan hides their latency.
        int pA = (nbase + 32 + l16) & (N_CODES - 1);
        int pB = (nbase + 48 + l16) & (N_CODES - 1);
        float nAx = cpos[3*pA+0], nAy = cpos[3*pA+1], nAz = cpos[3*pA+2];
        float nBx = cpos[3*pB+0], nBy = cpos[3*pB+1], nBz = cpos[3*pB+2];

        // B fragments (4x16): lanes 0-15 hold (-2cx,-2cy), lanes 16-31 (-2cz, 1).
        // C = |c|^2 broadcast per column => one WMMA = full d^2 tile.
        float cnA = cAx*cAx + cAy*cAy + cAz*cAz;
        float cnB = cBx*cBx + cBy*cBy + cBz*cBz;
        v2f b0, b1;
        b0.x = half ? (-2.0f*cAz) : (-2.0f*cAx);
        b0.y = half ? 1.0f        : (-2.0f*cAy);
        b1.x = half ? (-2.0f*cBz) : (-2.0f*cBx);
        b1.y = half ? 1.0f        : (-2.0f*cBy);
        v8f c0, c1;
        #pragma unroll
        for (int r = 0; r < 8; ++r) { c0[r] = cnA; c1[r] = cnB; }

        v8f d00 = __builtin_amdgcn_wmma_f32_16x16x4_f32(
            false, a0, false, b0, (short)0, c0, false, false);
        v8f d10 = __builtin_amdgcn_wmma_f32_16x16x4_f32(
            false, a1, false, b0, (short)0, c0, false, false);
        v8f d01 = __builtin_amdgcn_wmma_f32_16x16x4_f32(
            false, a0, false, b1, (short)0, c1, false, false);
        v8f d11 = __builtin_amdgcn_wmma_f32_16x16x4_f32(
            false, a1, false, b1, (short)0, c1, false, false);

        // Transpose through LDS: D element (M = r + 8*half, N = l16)
        #pragma unroll
        for (int r = 0; r < 8; ++r) {
            tileD[wave][     r + 8*half][     l16] = d00[r];
            tileD[wave][16 + r + 8*half][     l16] = d10[r];
            tileD[wave][     r + 8*half][16 + l16] = d01[r];
            tileD[wave][16 + r + 8*half][16 + l16] = d11[r];
        }
        __builtin_amdgcn_wave_barrier();   // writes before cross-lane reads (same wave)

        // ---- selection: lane owns query (qbase+lane), scans 32 candidates ----
        #pragma unroll 4
        for (int j = 0; j < 32; ++j) {
            float dv = tileD[wave][lane][j];
            if (dv < curMax) {
                listD[wave][lane][curMaxPos] = dv;
                listI[wave][lane][curMaxPos] = nbase + j;
                float m = -__builtin_inff(); int mp = 0;
                #pragma unroll
                for (int k = 0; k < KNN; ++k) {
                    float v = listD[wave][lane][k];
                    if (v > m) { m = v; mp = k; }
                }
                curMax = m; curMaxPos = mp;
            }
        }
        __builtin_amdgcn_wave_barrier();   // reads before next iteration's stores

        cAx = nAx; cAy = nAy; cAz = nAz;
        cBx = nBx; cBy = nBy; cBz = nBz;
    }

    // ---- weights: w = 1/(d^2 + eps) (DIST_SCALE=2), normalized in-place ----
    {
        float wsum = 0.0f;
        #pragma unroll
        for (int k = 0; k < KNN; ++k) {
            float dv = fmaxf(listD[wave][lane][k], 0.0f) + EPSF; // clamp fp cancellation
            float w = 1.0f / dv;
            listD[wave][lane][k] = w;
            wsum += w;
        }
        float inv = 1.0f / wsum;
        #pragma unroll
        for (int k = 0; k < KNN; ++k) listD[wave][lane][k] *= inv;
    }
    __builtin_amdgcn_wave_barrier();       // lane-private writes before wave-wide reads

    // ---- gather + weighted sum: wave walks its 32 queries, lane covers 4 dims ----
    // codes rows are 512B-aligned; lane*16B offsets keep float4 alignment; the
    // whole 4MB codes array is L2-resident (192MB L2), so gathers are L2-speed.
    for (int ql = 0; ql < 32; ++ql) {
        float4 acc = make_float4(0.f, 0.f, 0.f, 0.f);
        #pragma unroll 8
        for (int k = 0; k < KNN; ++k) {
            float w  = listD[wave][ql][k];            // broadcast LDS read
            int  idx = listI[wave][ql][k];
            const float4 cv =
                *reinterpret_cast<const float4*>(codes + idx * DIMS + lane * 4);
            acc.x += w * cv.x; acc.y += w * cv.y;
            acc.z += w * cv.z; acc.w += w * cv.w;
        }
        *reinterpret_cast<float4*>(out + (qbase + ql) * DIMS + lane * 4) = acc;
    }
}

extern "C" void kernel_launch(void* const* d_in, const int* in_sizes, int n_in,
                              void* d_out, int out_size, void* d_ws, size_t ws_size,
                              hipStream_t stream) {
    (void)in_sizes; (void)n_in; (void)d_ws; (void)ws_size; (void)out_size;
    // setup_inputs order: [0] indices (int64, unused: B=1, indices=[0]),
    // [1] query_points f32 (1,P,3), [2] codes_position f32 (1,N,3),
    // [3] codes f32 (1,N,D). Output: (P,D) f32.
    const float* qpos  = (const float*)d_in[1];
    const float* cpos  = (const float*)d_in[2];
    const float* codes = (const float*)d_in[3];
    float* outp = (float*)d_out;

    dim3 grid(P_TOTAL / 128);
    dim3 block(128);
    hipLaunchKernelGGL(CodeCloud_knn_wmma_kernel, grid, block, 0, stream,
                       qpos, cpos, codes, outp);
}